// ResidualGridVolume_61993557950730
// MI455X (gfx1250) — compile-verified
//
#include <hip/hip_runtime.h>
#include <cmath>

#define SCENE_BOUND 1.5f
#define MAX_C 32           // lane-per-channel requires C <= 32
#define WAVES_PER_BLOCK 8  // 256 threads = 8 wave32

// ---------------------------------------------------------------------------
// Prep: fused residual add + CDHW -> DHWC transpose into workspace.
// One wave handles 32 consecutive voxels. Source reads are coalesced and
// non-temporal (read-once streams, keep L2 for the staged volume); a
// wave-local LDS transpose makes the global stores fully coalesced float4s.
// ---------------------------------------------------------------------------
template <int C>
__global__ void prep_dhwc_fast(const float* __restrict__ base_d,
                               const float* __restrict__ base_sh,
                               const float* __restrict__ det_d,
                               const float* __restrict__ det_sh,
                               float* __restrict__ vol,   // [V][C]
                               int V)
{
    __shared__ float lds[WAVES_PER_BLOCK * 32 * C];

    const int wave = threadIdx.x >> 5;
    const int lane = threadIdx.x & 31;
    const int vbase = (blockIdx.x * WAVES_PER_BLOCK + wave) << 5; // *32
    const int v = vbase + lane;

    float* wlds = &lds[wave * 32 * C]; // voxel-major within wave: [lane][c]

    if (v < V) {
        wlds[lane * C + 0] = __builtin_nontemporal_load(&base_d[v]) +
                             __builtin_nontemporal_load(&det_d[v]);
        #pragma unroll
        for (int c = 1; c < C; ++c) {
            const size_t off = (size_t)(c - 1) * V + v; // coalesced across lanes
            wlds[lane * C + c] = __builtin_nontemporal_load(&base_sh[off]) +
                                 __builtin_nontemporal_load(&det_sh[off]);
        }
    }
    __syncthreads();

    if (vbase < V) {
        if (vbase + 32 <= V) {
            constexpr int n4 = (32 * C) >> 2;
            const float4* s4 = (const float4*)wlds;
            float4* d4 = (float4*)(vol + (size_t)vbase * C);
            #pragma unroll
            for (int i = 0; i < n4 / 32; ++i) d4[i * 32 + lane] = s4[i * 32 + lane];
        } else {
            const int tot = (V - vbase) * C;
            float* dst = vol + (size_t)vbase * C;
            for (int i = lane; i < tot; i += 32) dst[i] = wlds[i];
        }
    }
}

// ---------------------------------------------------------------------------
// Sampler: trilinear sample from DHWC volume, C and R compile-time.
// One wave per point; lane = channel. Interior points (the whole dataset,
// since |g|<=0.99) take a fast path: ONE base address + 8 immediate-offset
// global_load_b32 (all offsets fit the signed 24-bit IOFFSET), no clamps.
// The interior test is wave-uniform. Output stores are non-temporal so the
// 224MB write-once stream doesn't evict the volume from the 192MB L2.
// ---------------------------------------------------------------------------
template <int C, int R>
__global__ void sample_dhwc_fast(const float* __restrict__ xyz,
                                 const float* __restrict__ vol, // [V][C]
                                 float* __restrict__ out,       // [N][C]
                                 int N)
{
    const int lane = threadIdx.x & 31;
    const int wib = __builtin_amdgcn_readfirstlane((int)(threadIdx.x >> 5));
    const long long n = (long long)blockIdx.x * WAVES_PER_BLOCK + wib;
    if (n >= N) return;

    const float* p = xyz + 3ll * n; // uniform address -> s_load
    const float gx = p[0] / SCENE_BOUND;
    const float gy = p[1] / SCENE_BOUND;
    const float gz = p[2] / SCENE_BOUND;

    // (g+1)*0.5*(R-1): *0.5 is exact, fold into one exact constant.
    constexpr float s = 0.5f * (float)(R - 1);
    const float px = (gx + 1.0f) * s;
    const float py = (gy + 1.0f) * s;
    const float pz = (gz + 1.0f) * s;

    const float fx0 = floorf(px), fy0 = floorf(py), fz0 = floorf(pz);
    const float fx = px - fx0, fy = py - fy0, fz = pz - fz0;
    const int x0 = (int)fx0, y0 = (int)fy0, z0 = (int)fz0;

    const float wx0 = 1.0f - fx, wy0 = 1.0f - fy, wz0 = 1.0f - fz;

    const bool ch_ok = lane < C;
    float acc = 0.0f;

    const bool interior = (x0 >= 0) & (x0 <= R - 2) &
                          (y0 >= 0) & (y0 <= R - 2) &
                          (z0 >= 0) & (z0 <= R - 2); // wave-uniform

    if (interior) {
        const size_t v000 = ((size_t)z0 * R + y0) * R + x0;
        const float* pb = vol + v000 * C + lane;
        if (ch_ok) {
            // 8 loads, one address register, constant IOFFSETs (max ~1.85MB)
            const float f000 = pb[0];
            const float f001 = pb[C];
            const float f010 = pb[R * C];
            const float f011 = pb[R * C + C];
            const float f100 = pb[(size_t)R * R * C];
            const float f101 = pb[(size_t)R * R * C + C];
            const float f110 = pb[(size_t)R * R * C + R * C];
            const float f111 = pb[(size_t)R * R * C + R * C + C];
            // Reference accumulation order: dz outer, dy, dx inner.
            acc = fmaf(wx0 * wy0 * wz0, f000, acc);
            acc = fmaf(fx  * wy0 * wz0, f001, acc);
            acc = fmaf(wx0 * fy  * wz0, f010, acc);
            acc = fmaf(fx  * fy  * wz0, f011, acc);
            acc = fmaf(wx0 * wy0 * fz , f100, acc);
            acc = fmaf(fx  * wy0 * fz , f101, acc);
            acc = fmaf(wx0 * fy  * fz , f110, acc);
            acc = fmaf(fx  * fy  * fz , f111, acc);
        }
    } else {
        #pragma unroll
        for (int dz = 0; dz < 2; ++dz)
        #pragma unroll
        for (int dy = 0; dy < 2; ++dy)
        #pragma unroll
        for (int dx = 0; dx < 2; ++dx) {
            const int xi = x0 + dx, yi = y0 + dy, zi = z0 + dz;
            const bool inb = (xi >= 0) & (xi < R) & (yi >= 0) & (yi < R) &
                             (zi >= 0) & (zi < R);
            const int xc = min(max(xi, 0), R - 1);
            const int yc = min(max(yi, 0), R - 1);
            const int zc = min(max(zi, 0), R - 1);
            float w = (dx ? fx : wx0) * (dy ? fy : wy0);
            w *= (dz ? fz : wz0);
            w = inb ? w : 0.0f;
            const size_t vox = ((size_t)zc * R + yc) * R + xc;
            const float feat = ch_ok ? vol[vox * C + lane] : 0.0f;
            acc = fmaf(w, feat, acc);
        }
    }

    if (ch_ok) __builtin_nontemporal_store(acc, &out[(size_t)n * C + lane]);
}

// ---------------------------------------------------------------------------
// Generic (runtime C/R) versions — used only if shapes differ from 28/128.
// ---------------------------------------------------------------------------
__global__ void prep_dhwc_kernel(const float* __restrict__ base_d,
                                 const float* __restrict__ base_sh,
                                 const float* __restrict__ det_d,
                                 const float* __restrict__ det_sh,
                                 float* __restrict__ vol, int V, int C)
{
    __shared__ float lds[WAVES_PER_BLOCK * 32 * MAX_C];
    const int wave = threadIdx.x >> 5;
    const int lane = threadIdx.x & 31;
    const int vbase = (blockIdx.x * WAVES_PER_BLOCK + wave) << 5;
    const int v = vbase + lane;
    float* wlds = &lds[wave * 32 * C];

    if (v < V) {
        wlds[lane * C + 0] = base_d[v] + det_d[v];
        for (int c = 1; c < C; ++c) {
            size_t off = (size_t)(c - 1) * V + v;
            wlds[lane * C + c] = base_sh[off] + det_sh[off];
        }
    }
    __syncthreads();
    if (vbase < V) {
        const int nval = min(32, V - vbase);
        const int tot = nval * C;
        float* dst = vol + (size_t)vbase * C;
        for (int i = lane; i < tot; i += 32) dst[i] = wlds[i];
    }
}

__global__ void sample_dhwc_kernel(const float* __restrict__ xyz,
                                   const float* __restrict__ vol,
                                   float* __restrict__ out,
                                   int N, int R, int C)
{
    const int lane = threadIdx.x & 31;
    const int wib = __builtin_amdgcn_readfirstlane((int)(threadIdx.x >> 5));
    const long long n = (long long)blockIdx.x * WAVES_PER_BLOCK + wib;
    if (n >= N) return;

    const float* p = xyz + 3ll * n;
    const float gx = p[0] / SCENE_BOUND;
    const float gy = p[1] / SCENE_BOUND;
    const float gz = p[2] / SCENE_BOUND;
    const float s = 0.5f * (float)(R - 1);
    const float px = (gx + 1.0f) * s, py = (gy + 1.0f) * s, pz = (gz + 1.0f) * s;
    const float fx0 = floorf(px), fy0 = floorf(py), fz0 = floorf(pz);
    const float fx = px - fx0, fy = py - fy0, fz = pz - fz0;
    const int x0 = (int)fx0, y0 = (int)fy0, z0 = (int)fz0;

    const bool ch_ok = lane < C;
    float acc = 0.0f;
    #pragma unroll
    for (int dz = 0; dz < 2; ++dz)
    #pragma unroll
    for (int dy = 0; dy < 2; ++dy)
    #pragma unroll
    for (int dx = 0; dx < 2; ++dx) {
        const int xi = x0 + dx, yi = y0 + dy, zi = z0 + dz;
        const bool inb = (xi >= 0) & (xi < R) & (yi >= 0) & (yi < R) &
                         (zi >= 0) & (zi < R);
        const int xc = min(max(xi, 0), R - 1);
        const int yc = min(max(yi, 0), R - 1);
        const int zc = min(max(zi, 0), R - 1);
        float w = (dx ? fx : 1.0f - fx) * (dy ? fy : 1.0f - fy);
        w *= (dz ? fz : 1.0f - fz);
        w = inb ? w : 0.0f;
        const size_t vox = ((size_t)zc * R + yc) * R + xc;
        const float feat = ch_ok ? vol[vox * C + lane] : 0.0f;
        acc = fmaf(w, feat, acc);
    }
    if (ch_ok) out[(size_t)n * C + lane] = acc;
}

__global__ void sample_direct_kernel(const float* __restrict__ xyz,
                                     const float* __restrict__ base_d,
                                     const float* __restrict__ base_sh,
                                     const float* __restrict__ det_d,
                                     const float* __restrict__ det_sh,
                                     float* __restrict__ out,
                                     int N, int R, int C)
{
    const int lane = threadIdx.x & 31;
    const int wib = __builtin_amdgcn_readfirstlane((int)(threadIdx.x >> 5));
    const long long n = (long long)blockIdx.x * WAVES_PER_BLOCK + wib;
    if (n >= N) return;

    const size_t V = (size_t)R * R * R;
    const float* p = xyz + 3ll * n;
    const float gx = p[0] / SCENE_BOUND;
    const float gy = p[1] / SCENE_BOUND;
    const float gz = p[2] / SCENE_BOUND;
    const float s = 0.5f * (float)(R - 1);
    const float px = (gx + 1.0f) * s, py = (gy + 1.0f) * s, pz = (gz + 1.0f) * s;
    const float fx0 = floorf(px), fy0 = floorf(py), fz0 = floorf(pz);
    const float fx = px - fx0, fy = py - fy0, fz = pz - fz0;
    const int x0 = (int)fx0, y0 = (int)fy0, z0 = (int)fz0;

    const bool ch_ok = lane < C;
    float acc = 0.0f;
    #pragma unroll
    for (int dz = 0; dz < 2; ++dz)
    #pragma unroll
    for (int dy = 0; dy < 2; ++dy)
    #pragma unroll
    for (int dx = 0; dx < 2; ++dx) {
        const int xi = x0 + dx, yi = y0 + dy, zi = z0 + dz;
        const bool inb = (xi >= 0) & (xi < R) & (yi >= 0) & (yi < R) &
                         (zi >= 0) & (zi < R);
        const int xc = min(max(xi, 0), R - 1);
        const int yc = min(max(yi, 0), R - 1);
        const int zc = min(max(zi, 0), R - 1);
        float w = (dx ? fx : 1.0f - fx) * (dy ? fy : 1.0f - fy);
        w *= (dz ? fz : 1.0f - fz);
        w = inb ? w : 0.0f;
        const size_t vox = ((size_t)zc * R + yc) * R + xc;
        float feat = 0.0f;
        if (ch_ok) {
            if (lane == 0) feat = base_d[vox] + det_d[vox];
            else {
                size_t off = (size_t)(lane - 1) * V + vox;
                feat = base_sh[off] + det_sh[off];
            }
        }
        acc = fmaf(w, feat, acc);
    }
    if (ch_ok) out[(size_t)n * C + lane] = acc;
}

extern "C" void kernel_launch(void* const* d_in, const int* in_sizes, int n_in,
                              void* d_out, int out_size, void* d_ws, size_t ws_size,
                              hipStream_t stream) {
    const float* xyz     = (const float*)d_in[0];
    const float* base_d  = (const float*)d_in[1];
    const float* base_sh = (const float*)d_in[2];
    const float* det_d   = (const float*)d_in[3];
    const float* det_sh  = (const float*)d_in[4];
    float* out = (float*)d_out;

    const int N = in_sizes[0] / 3;
    const long long V = in_sizes[1];                 // R^3
    int R = (int)llround(cbrt((double)V));
    while ((long long)R * R * R > V) --R;
    while ((long long)(R + 1) * (R + 1) * (R + 1) <= V) ++R;
    const int C = 1 + (int)(in_sizes[2] / V);        // 1 density + SH channels

    const size_t need = (size_t)V * (size_t)C * sizeof(float);
    const int sample_blocks =
        (int)(((long long)N + WAVES_PER_BLOCK - 1) / WAVES_PER_BLOCK);
    const long long nwaves = (V + 31) / 32;
    const int prep_blocks =
        (int)((nwaves + WAVES_PER_BLOCK - 1) / WAVES_PER_BLOCK);

    if (C == 28 && R == 128 && ws_size >= need) {
        float* vol = (float*)d_ws;
        prep_dhwc_fast<28><<<prep_blocks, 32 * WAVES_PER_BLOCK, 0, stream>>>(
            base_d, base_sh, det_d, det_sh, vol, (int)V);
        sample_dhwc_fast<28, 128><<<sample_blocks, 32 * WAVES_PER_BLOCK, 0, stream>>>(
            xyz, vol, out, N);
    } else if (C <= MAX_C && ws_size >= need) {
        float* vol = (float*)d_ws;
        prep_dhwc_kernel<<<prep_blocks, 32 * WAVES_PER_BLOCK, 0, stream>>>(
            base_d, base_sh, det_d, det_sh, vol, (int)V, C);
        sample_dhwc_kernel<<<sample_blocks, 32 * WAVES_PER_BLOCK, 0, stream>>>(
            xyz, vol, out, N, R, C);
    } else {
        sample_direct_kernel<<<sample_blocks, 32 * WAVES_PER_BLOCK, 0, stream>>>(
            xyz, base_d, base_sh, det_d, det_sh, out, N, R, C);
    }
}